// LengthRegulator_34351148433503
// MI455X (gfx1250) — compile-verified
//
#include <hip/hip_runtime.h>

typedef float v2f __attribute__((ext_vector_type(2)));
typedef float v8f __attribute__((ext_vector_type(8)));

#define NB    16
#define LSEQ  512
#define CH    256     // D == F == 256
#define TMAX  4096
#define LN_EPS 1e-5f

// ---------------------------------------------------------------------------
// Fused conv1d(K=3, same-pad) + bias + LayerNorm + ReLU (+ optional F->1
// linear + ReLU producing dpo). GEMM view: [NB*LSEQ, 3*256] x [3*256, 256].
// One block = 32 rows x 256 cols; 8 waves, each wave owns 2 row-tiles x
// 2 col-tiles (four 16x16 tiles), accumulated with V_WMMA_F32_16X16X4_F32.
// fp32 end to end: the result feeds int(dpo+0.5), so no 16-bit rounding.
// A-edge handling: unconditional load from a clamped row + v_cndmask zeroing
// (no EXEC divergence in the hot loop).
// ---------------------------------------------------------------------------
__global__ void __launch_bounds__(256)
conv_ln_kernel(const float* __restrict__ xin,    // [NB, LSEQ, CH]
               const float* __restrict__ wgt,    // [3, CH, CH]  (k, cin, cout)
               const float* __restrict__ bias,   // [CH]
               const float* __restrict__ gam,    // [CH]
               const float* __restrict__ bet,    // [CH]
               const float* __restrict__ lw,     // [CH] or nullptr
               const float* __restrict__ lb,     // [1]  or nullptr
               float* __restrict__ h_out,        // [NB*LSEQ, CH] or nullptr
               float* __restrict__ dpo_out,      // [NB*LSEQ]     or nullptr
               int do_linear)
{
    const int tid   = threadIdx.x;
    const int wave  = tid >> 5;
    const int lane  = tid & 31;
    const int half  = lane >> 4;       // 0: K pair {0,1}, 1: K pair {2,3}
    const int m16   = lane & 15;       // M for A-frag / N for B,C,D frags
    const int m0    = blockIdx.x * 32; // global row base (n*512 + l)
    const int n     = m0 >> 9;
    const int lbase = m0 & (LSEQ - 1);
    const int fbase = wave * 32;       // this wave's column base (2 tiles)
    const int kbase = half * 2;

    v8f acc00 = {};   // row-tile 0, col-tile 0
    v8f acc01 = {};   // row-tile 0, col-tile 1
    v8f acc10 = {};   // row-tile 1, col-tile 0
    v8f acc11 = {};   // row-tile 1, col-tile 1

    const int lrow0 = lbase + m16;        // A-row positions for the 2 tiles
    const int lrow1 = lbase + 16 + m16;

    for (int k = 0; k < 3; ++k) {
        const int  lp0 = lrow0 + k - 1;
        const int  lp1 = lrow1 + k - 1;
        const bool v0  = (lp0 >= 0) && (lp0 < LSEQ);
        const bool v1  = (lp1 >= 0) && (lp1 < LSEQ);
        const int  cl0 = lp0 < 0 ? 0 : (lp0 > LSEQ - 1 ? LSEQ - 1 : lp0);
        const int  cl1 = lp1 < 0 ? 0 : (lp1 > LSEQ - 1 ? LSEQ - 1 : lp1);
        const float* __restrict__ arow0 = xin + (size_t)(n * LSEQ + cl0) * CH;
        const float* __restrict__ arow1 = xin + (size_t)(n * LSEQ + cl1) * CH;
        const float* __restrict__ wk    = wgt + (size_t)k * CH * CH;

        #pragma unroll 4
        for (int kc = 0; kc < CH; kc += 4) {
            const int ch = kc + kbase;

            // Unconditional b64 loads from clamped rows, then cndmask-zero.
            v2f a0r = *(const v2f*)(arow0 + ch);
            v2f a1r = *(const v2f*)(arow1 + ch);
            v2f a0, a1;
            a0[0] = v0 ? a0r[0] : 0.0f;  a0[1] = v0 ? a0r[1] : 0.0f;
            a1[0] = v1 ? a1r[0] : 0.0f;  a1[1] = v1 ? a1r[1] : 0.0f;

            const float* __restrict__ wp = wk + (size_t)ch * CH + fbase + m16;
            v2f b0, b1;
            b0[0] = wp[0];   b0[1] = wp[CH];
            b1[0] = wp[16];  b1[1] = wp[CH + 16];

            acc00 = __builtin_amdgcn_wmma_f32_16x16x4_f32(
                false, a0, false, b0, (short)0, acc00, false, false);
            acc01 = __builtin_amdgcn_wmma_f32_16x16x4_f32(
                false, a0, false, b1, (short)0, acc01, false, false);
            acc10 = __builtin_amdgcn_wmma_f32_16x16x4_f32(
                false, a1, false, b0, (short)0, acc10, false, false);
            acc11 = __builtin_amdgcn_wmma_f32_16x16x4_f32(
                false, a1, false, b1, (short)0, acc11, false, false);
        }
    }

    // Stage the 32x256 block (with bias) to LDS for row-wise LayerNorm.
    __shared__ float tile[32][CH + 8];   // +8 floats pad vs 64-bank conflicts
    const int c0 = fbase + m16;
    #pragma unroll
    for (int r = 0; r < 8; ++r) {
        const int rowi = r + half * 8;   // D-frag: VGPR r -> M = r + half*8
        tile[rowi][c0]           = acc00[r] + bias[c0];
        tile[rowi][c0 + 16]      = acc01[r] + bias[c0 + 16];
        tile[rowi + 16][c0]      = acc10[r] + bias[c0];
        tile[rowi + 16][c0 + 16] = acc11[r] + bias[c0 + 16];
    }
    __syncthreads();

    // LayerNorm + ReLU: wave w handles rows 4w .. 4w+3 (wave32 shuffles).
    for (int rr = wave * 4; rr < wave * 4 + 4; ++rr) {
        float v[8];
        float s = 0.0f, s2 = 0.0f;
        #pragma unroll
        for (int j = 0; j < 8; ++j) {
            const float t = tile[rr][lane + 32 * j];
            v[j] = t; s += t; s2 += t * t;
        }
        #pragma unroll
        for (int off = 16; off > 0; off >>= 1) {
            s  += __shfl_xor(s,  off, 32);
            s2 += __shfl_xor(s2, off, 32);
        }
        const float mu   = s * (1.0f / CH);
        const float varr = s2 * (1.0f / CH) - mu * mu;
        const float rstd = rsqrtf(varr + LN_EPS);
        const int   grow = m0 + rr;

        if (do_linear) {
            float dot = 0.0f;
            #pragma unroll
            for (int j = 0; j < 8; ++j) {
                const int c = lane + 32 * j;
                const float h = fmaxf((v[j] - mu) * rstd * gam[c] + bet[c], 0.0f);
                dot += h * lw[c];
            }
            #pragma unroll
            for (int off = 16; off > 0; off >>= 1)
                dot += __shfl_xor(dot, off, 32);
            if (lane == 0)
                dpo_out[grow] = fmaxf(dot + lb[0], 0.0f);
        } else {
            #pragma unroll
            for (int j = 0; j < 8; ++j) {
                const int c = lane + 32 * j;
                h_out[(size_t)grow * CH + c] =
                    fmaxf((v[j] - mu) * rstd * gam[c] + bet[c], 0.0f);
            }
        }
    }
}

// ---------------------------------------------------------------------------
// Durations -> exclusive scan -> interval map. map[n,t] = token index l whose
// [start,end) covers t, else -1. Intervals are disjoint => race-free fill.
// ---------------------------------------------------------------------------
__global__ void __launch_bounds__(512)
durmap_kernel(const float* __restrict__ dpo, int* __restrict__ map)
{
    const int n = blockIdx.x;
    const int l = threadIdx.x;
    __shared__ int sdur[LSEQ];

    const int dur = (int)(dpo[n * LSEQ + l] + 0.5f);  // ALPHA == 1.0
    sdur[l] = dur;
    for (int t = l; t < TMAX; t += LSEQ)
        map[n * TMAX + t] = -1;
    __syncthreads();

    if (threadIdx.x == 0) {               // 512 serial adds: negligible
        int acc = 0;
        for (int i = 0; i < LSEQ; ++i) { const int d = sdur[i]; sdur[i] = acc; acc += d; }
    }
    __syncthreads();

    int start = sdur[l];
    int end   = start + dur;
    if (start > TMAX) start = TMAX;
    if (end   > TMAX) end   = TMAX;
    for (int t = start; t < end; ++t)
        map[n * TMAX + t] = l;
}

// ---------------------------------------------------------------------------
// Bandwidth stage: out[n,t,:] = x[n, map[n,t], :] or zeros. float4 rows.
// 64 MB of stores at 23.3 TB/s dominates the whole pipeline; x (8 MB) is
// L2-resident so the gather reads come from L2.
// ---------------------------------------------------------------------------
__global__ void __launch_bounds__(256)
gather_kernel(const float* __restrict__ x, const int* __restrict__ map,
              float* __restrict__ out)
{
    const int sub    = threadIdx.x >> 6;        // 4 rows per block
    const int lane64 = threadIdx.x & 63;        // 64 * float4 = 256 floats
    const int row    = blockIdx.x * 4 + sub;    // n*TMAX + t
    const int n      = row >> 12;
    const int l      = map[row];

    float4 v = make_float4(0.f, 0.f, 0.f, 0.f);
    if (l >= 0)
        v = ((const float4*)(x + (size_t)(n * LSEQ + l) * CH))[lane64];
    ((float4*)(out + (size_t)row * CH))[lane64] = v;
}

__global__ void __launch_bounds__(256)
pos_kernel(int* __restrict__ pos)
{
    const int t = blockIdx.x * 256 + threadIdx.x;
    if (t < TMAX) pos[t] = t + 1;
}

// ---------------------------------------------------------------------------
extern "C" void kernel_launch(void* const* d_in, const int* in_sizes, int n_in,
                              void* d_out, int out_size, void* d_ws, size_t ws_size,
                              hipStream_t stream)
{
    const float* x       = (const float*)d_in[0];
    const float* conv1_w = (const float*)d_in[1];
    const float* conv1_b = (const float*)d_in[2];
    const float* ln1_g   = (const float*)d_in[3];
    const float* ln1_b   = (const float*)d_in[4];
    const float* conv2_w = (const float*)d_in[5];
    const float* conv2_b = (const float*)d_in[6];
    const float* ln2_g   = (const float*)d_in[7];
    const float* ln2_b   = (const float*)d_in[8];
    const float* lin_w   = (const float*)d_in[9];
    const float* lin_b   = (const float*)d_in[10];

    // Workspace layout: h1 (8 MB) | dpo (32 KB) | map (256 KB)
    float* h1  = (float*)d_ws;
    float* dpo = h1 + (size_t)NB * LSEQ * CH;
    int*   map = (int*)(dpo + (size_t)NB * LSEQ);

    float* out = (float*)d_out;                               // [NB, TMAX, CH]
    int*   pos = (int*)(out + (size_t)NB * TMAX * CH);        // [TMAX] int32

    const int rowBlocks = (NB * LSEQ) / 32;                   // 256 blocks

    conv_ln_kernel<<<rowBlocks, 256, 0, stream>>>(
        x, conv1_w, conv1_b, ln1_g, ln1_b,
        nullptr, nullptr, h1, nullptr, 0);

    conv_ln_kernel<<<rowBlocks, 256, 0, stream>>>(
        h1, conv2_w, conv2_b, ln2_g, ln2_b,
        lin_w, lin_b, nullptr, dpo, 1);

    durmap_kernel<<<NB, LSEQ, 0, stream>>>(dpo, map);

    gather_kernel<<<(NB * TMAX) / 4, 256, 0, stream>>>(x, map, out);

    pos_kernel<<<TMAX / 256, 256, 0, stream>>>(pos);
}